// RoiAlign_2705829396905
// MI455X (gfx1250) — compile-verified
//
#include <hip/hip_runtime.h>
#include <cstdint>

// RoiAlign (TF crop_and_resize, POOL=7) for MI455X gfx1250.
// Memory-bound kernel: 205MB NT-streamed output, L2-resident feature map,
// b128 coalesced gathers, async global->LDS box broadcast, global_prefetch.

typedef float v4f __attribute__((ext_vector_type(4)));
typedef int   v4i __attribute__((ext_vector_type(4)));

// Address-space-qualified int4 for the gfx1250 async LDS-DMA builtin.
typedef __attribute__((address_space(1))) v4i gv4i;   // global
typedef __attribute__((address_space(3))) v4i lv4i;   // LDS

#define FM_H   64
#define FM_W   64
#define FM_C   256
#define NBOX   512
#define NPIX   49   // 7*7

__global__ __launch_bounds__(256)
void roi_align_kernel(const float* __restrict__ fm,
                      const float* __restrict__ rpn,
                      float* __restrict__ out)
{
    const int bn  = blockIdx.x;           // b*NBOX + n
    const int b   = bn >> 9;              // NBOX == 512
    const int tid = threadIdx.x;
    const int c4  = (tid & 63) << 2;      // 4-channel base for this lane (16B)
    const int grp = tid >> 6;             // pixel group 0..3

    __shared__ v4f s_box;

    const float* boxp = rpn + (size_t)bn * 4;

#if __has_builtin(__builtin_amdgcn_global_load_async_to_lds_b128)
    // CDNA5 async copy path (ASYNCcnt): one 16B global->LDS DMA for the
    // block-uniform box instead of 256 redundant per-lane loads.
    if (tid == 0) {
        __builtin_amdgcn_global_load_async_to_lds_b128(
            (gv4i*)(uintptr_t)boxp,
            (lv4i*)(uintptr_t)&s_box,
            0, 0);
    }
  #if __has_builtin(__builtin_amdgcn_s_wait_asynccnt)
    __builtin_amdgcn_s_wait_asynccnt(0);
  #else
    asm volatile("s_wait_asynccnt 0" ::: "memory");
  #endif
#else
    if (tid == 0) s_box = *(const v4f*)boxp;
#endif
    __syncthreads();

    // boxes are (x1,y1,x2,y2), normalized, x1<=x2, y1<=y2 by construction.
    const float x1 = s_box.x, y1 = s_box.y, x2 = s_box.z, y2 = s_box.w;
    // Match reference expression order exactly:
    //   v = v1*(S-1) + i * ((v2-v1)*(S-1)/(POOL-1))
    const float by = y1 * 63.0f;
    const float bx = x1 * 63.0f;
    const float sy = (y2 - y1) * 63.0f / 6.0f;
    const float sx = (x2 - x1) * 63.0f / 6.0f;

    const float* fmb  = fm + (size_t)b * (FM_H * FM_W * FM_C);
    float*       outb = out + (size_t)bn * (NPIX * FM_C);

    for (int p = grp; p < NPIX; p += 4) {
        const int py = p / 7;
        const int px = p - py * 7;

        const float yv  = by + (float)py * sy;
        const float xv  = bx + (float)px * sx;
        const float y0f = floorf(yv);
        const float x0f = floorf(xv);
        const float fy  = yv - y0f;
        const float fx  = xv - x0f;

        int iy0 = (int)y0f; iy0 = iy0 < 0 ? 0 : (iy0 > 63 ? 63 : iy0);
        int ix0 = (int)x0f; ix0 = ix0 < 0 ? 0 : (ix0 > 63 ? 63 : ix0);
        const int iy1 = (iy0 + 1 > 63) ? 63 : iy0 + 1;
        const int ix1 = (ix0 + 1 > 63) ? 63 : ix0 + 1;

        // (y*64 + x) * 256  ==  ((y<<6)+x)<<8 ; channels contiguous -> b128/lane,
        // 64 lanes cover a fully coalesced 1KB span per corner.
        const int o00 = ((iy0 << 6) + ix0) << 8;
        const int o01 = ((iy0 << 6) + ix1) << 8;
        const int o10 = ((iy1 << 6) + ix0) << 8;
        const int o11 = ((iy1 << 6) + ix1) << 8;

        const v4f tl = *(const v4f*)(fmb + o00 + c4);
        const v4f tr = *(const v4f*)(fmb + o01 + c4);
        const v4f bl = *(const v4f*)(fmb + o10 + c4);
        const v4f br = *(const v4f*)(fmb + o11 + c4);

        const v4f top = tl + (tr - tl) * fx;
        const v4f bot = bl + (br - bl) * fx;
        const v4f res = top + (bot - top) * fy;

        // Streaming 205MB output: non-temporal so the L2 keeps the feature map.
        __builtin_nontemporal_store(res, (v4f*)(outb + (p << 8) + c4));

        // Prefetch next iteration's top-left corner row (global_prefetch_b8).
        const int pn = p + 4;
        if (pn < NPIX) {
            const int pyn = pn / 7;
            const int pxn = pn - pyn * 7;
            int iyn = (int)floorf(by + (float)pyn * sy);
            int ixn = (int)floorf(bx + (float)pxn * sx);
            iyn = iyn < 0 ? 0 : (iyn > 63 ? 63 : iyn);
            ixn = ixn < 0 ? 0 : (ixn > 63 ? 63 : ixn);
            __builtin_prefetch(fmb + ((((iyn << 6) + ixn) << 8) + c4), 0, 1);
        }
    }
}

extern "C" void kernel_launch(void* const* d_in, const int* in_sizes, int n_in,
                              void* d_out, int out_size, void* d_ws, size_t ws_size,
                              hipStream_t stream) {
    (void)in_sizes; (void)n_in; (void)d_ws; (void)ws_size; (void)out_size;
    const float* fm  = (const float*)d_in[0];   // [8,64,64,256] f32
    const float* rpn = (const float*)d_in[1];   // [8,512,4]     f32
    float* out = (float*)d_out;                 // [8,512,7,7,256] f32

    dim3 grid(8 * NBOX);   // one block per (image, box)
    dim3 block(256);       // 8 wave32 waves: 4 pixel-groups x 64 channel-lanes
    roi_align_kernel<<<grid, block, 0, stream>>>(fm, rpn, out);
}